// CIN_40269613368094
// MI455X (gfx1250) — compile-verified
//
#include <hip/hip_runtime.h>

// ---------------------------------------------------------------------------
// CIN (xDeepFM compressed-interaction) for MI455X / gfx1250, wave32 + WMMA.
//   B=4096, F=39, D=16, L0=128, H1=64, K0=F*F=1521, K1=H1*F=2496
// Per batch: Y0(128x16) = W0(128x1521) x Z0(1521x16),  relu+bias
//            Y1(128x16) = W1(128x2496) x Z1(2496x16),  relu+bias
//            out[b,0:64]   = sum_d Y0[64:128, d]
//            out[b,64:192] = sum_d Y1[0:128, d]
// Matrix op: V_WMMA_F32_16X16X4_F32 (fp32 precision parity with reference).
// B operands staged in LDS with a 2-step swizzle so one ds_load_b128 feeds
// two WMMAs per batch.  The layer-0 tail chunk (the only one needing A-column
// clamping) is peeled out of the chunk loop so the loop body has a single
// chunk_gemm instantiation and the accumulators stay pinned across the
// backedge (avoids the 32x v_dual_mov phi-copy block seen previously).
// ---------------------------------------------------------------------------

typedef float v2f __attribute__((ext_vector_type(2)));
typedef float v4f __attribute__((ext_vector_type(4)));
typedef float v8f __attribute__((ext_vector_type(8)));

#define B_TOT 4096
#define F     39
#define D     16
#define L0N   128
#define H1    64
#define K0    1521
#define K1    2496
#define NB    4        // batches fused per block (amortizes W streaming from L2)
#define TPB   256      // 8 wave32 waves; wave w owns output rows [16w, 16w+16)
#define CK    156      // K-chunk = 4 h-rows * 39
#define ZSTRIDE 2560   // per-batch Z region (20 step-pairs * 128 floats)

// LDS layout (floats)
#define XS_OFF 0                     // xs: NB*F*D      = 2496
#define YS_OFF 2496                  // ys: NB*H1*D     = 4096 (relu'd y0 rows 0..63)
#define ZS_OFF (2496 + 4096)         // zs: NB*ZSTRIDE  = 10240 (Z chunk / reduce scratch)
#define LDS_FLOATS (2496 + 4096 + 10240)

// Two-step swizzle: element (c,d) with S=c>>3, t=(c>>2)&1, q=(c>>1)&1, e=c&1
// lives at S*128 + q*64 + d*4 + t*2 + e.  A lane (half q, col d) then reads
// one aligned float4 at S*128 + q*64 + d*4 holding the B pairs of wmma steps
// 2S (elements 0,1) and 2S+1 (elements 2,3).
__device__ __forceinline__ int zswz2(int c, int d) {
  return ((c >> 3) << 7) + (((c >> 1) & 1) << 6) + (d << 2) +
         (((c >> 2) & 1) << 1) + (c & 1);
}

// One K-chunk of the GEMM: `pairs` step-pairs (+ optional leftover step).
// wr = W row pointer for this lane (A row).  CLAMP: clamp A columns to
// Klen-1 (only needed for the zero-padded layer-0 tail; padded Z entries are
// zero so the clamped values never contribute).  AL64: row stride is even so
// the A pair can be one aligned 8-byte load.
template <bool CLAMP, bool AL64>
__device__ __forceinline__ void chunk_gemm(
    const float* __restrict__ wr, int Klen, int cbase, int pairs, bool tail,
    int hi, int boff, const float* __restrict__ zs, v8f (&acc)[NB])
{
  for (int S = 0; S < pairs; ++S) {
    const int cg = cbase + (S << 3) + 2 * hi;
    v2f a0, a1;
    if (AL64) {
      a0 = *(const v2f*)&wr[cg];
      a1 = *(const v2f*)&wr[cg + 4];
    } else if (CLAMP) {
      a0.x = wr[(cg     < Klen) ? cg     : Klen - 1];
      a0.y = wr[(cg + 1 < Klen) ? cg + 1 : Klen - 1];
      a1.x = wr[(cg + 4 < Klen) ? cg + 4 : Klen - 1];
      a1.y = wr[(cg + 5 < Klen) ? cg + 5 : Klen - 1];
    } else {
      a0.x = wr[cg];     a0.y = wr[cg + 1];
      a1.x = wr[cg + 4]; a1.y = wr[cg + 5];
    }
    const int zoff = (S << 7) + boff;
#pragma unroll
    for (int nb = 0; nb < NB; ++nb) {
      v4f bq = *(const v4f*)&zs[nb * ZSTRIDE + zoff];   // ds_load_b128
      v2f bA = __builtin_shufflevector(bq, bq, 0, 1);
      v2f bB = __builtin_shufflevector(bq, bq, 2, 3);
      acc[nb] = __builtin_amdgcn_wmma_f32_16x16x4_f32(
          false, a0, false, bA, (short)0, acc[nb], false, false);
      acc[nb] = __builtin_amdgcn_wmma_f32_16x16x4_f32(
          false, a1, false, bB, (short)0, acc[nb], false, false);
    }
  }
  if (tail) {                         // leftover odd step s = 2*pairs
    const int s  = pairs << 1;
    const int cg = cbase + (s << 2) + 2 * hi;
    v2f a0;
    if (AL64) {
      a0 = *(const v2f*)&wr[cg];
    } else {
      a0.x = wr[cg]; a0.y = wr[cg + 1];
    }
    const int zoff = (pairs << 7) + boff;
#pragma unroll
    for (int nb = 0; nb < NB; ++nb) {
      v2f b = *(const v2f*)&zs[nb * ZSTRIDE + zoff];    // ds_load_b64
      acc[nb] = __builtin_amdgcn_wmma_f32_16x16x4_f32(
          false, a0, false, b, (short)0, acc[nb], false, false);
    }
  }
}

__global__ __launch_bounds__(TPB) void cin_wmma_kernel(
    const float* __restrict__ x,  const float* __restrict__ W0,
    const float* __restrict__ b0, const float* __restrict__ W1,
    const float* __restrict__ b1, float* __restrict__ out)
{
  __shared__ __align__(16) float lds[LDS_FLOATS];
  float* xs = lds + XS_OFF;
  float* ys = lds + YS_OFF;
  float* zs = lds + ZS_OFF;

  const int tid   = threadIdx.x;
  const int wave  = tid >> 5;       // 0..7 : M-tile (16 output rows)
  const int lane  = tid & 31;
  const int lo16  = lane & 15;      // N index (= d) for B/C/D operands
  const int hi    = lane >> 4;      // selects K pair {0,1} vs {2,3}
  const int bbase = blockIdx.x * NB;
  const int arow  = (wave << 4) + lo16;   // A-operand row (M = lane%16)
  const int boff  = (hi << 6) + (lo16 << 2);

  // ---- stage x for 4 consecutive batches (contiguous in global) ----
  const float* xg = x + (size_t)bbase * (F * D);
  for (int e = tid; e < NB * F * D; e += TPB) xs[e] = xg[e];

  v8f acc[NB];
#pragma unroll
  for (int nb = 0; nb < NB; ++nb)
#pragma unroll
    for (int j = 0; j < 8; ++j) acc[nb][j] = 0.0f;

  const float* w0r = W0 + (size_t)arow * K0;
  const float* w1r = W1 + (size_t)arow * K1;

  // ======================= layer 0 : K0 = 1521 ==========================
  // 9 uniform chunks (4 h-rows, 156 cols, 39 steps, clamp-free: max col 1403)
  // + one peeled tail chunk (3 h-rows, 117 cols -> pad 120, 30 steps, clamp).
  for (int ch = 0; ch < 9; ++ch) {
    const int h0    = ch << 2;
    const int cbase = ch * CK;

    __syncthreads();                       // previous chunk fully consumed
    // stage Z0 chunk: z[hh*39+mm, d] = x[h0+hh][d] * x[mm][d]
    for (int nb = 0; nb < NB; ++nb) {
      const float* xb = xs + nb * (F * D);
      float*       zb = zs + nb * ZSTRIDE;
      for (int hh = 0; hh < 4; ++hh) {
        for (int r = tid; r < F * D; r += TPB) {
          const int mm = r >> 4;
          const int d  = r & 15;
          zb[zswz2(hh * F + mm, d)] =
              xb[(h0 + hh) * D + d] * xb[mm * D + d];
        }
      }
    }
    __syncthreads();

    __builtin_prefetch(&w0r[cbase + CK + 2 * hi], 0, 0);   // next W0 chunk
    chunk_gemm<false, false>(w0r, K0, cbase, 19, true, hi, boff, zs, acc);
  }

  {  // ---- peeled layer-0 tail chunk: ch = 9 ----
    __syncthreads();
    for (int nb = 0; nb < NB; ++nb) {
      const float* xb = xs + nb * (F * D);
      float*       zb = zs + nb * ZSTRIDE;
      for (int hh = 0; hh < 3; ++hh) {
        for (int r = tid; r < F * D; r += TPB) {
          const int mm = r >> 4;
          const int d  = r & 15;
          zb[zswz2(hh * F + mm, d)] =
              xb[(36 + hh) * D + d] * xb[mm * D + d];
        }
      }
    }
    // zero-fill pad columns 117..119
    for (int idx = tid; idx < NB * 3 * D; idx += TPB) {
      const int nb = idx / (3 * D);
      const int rr = idx - nb * (3 * D);
      zs[nb * ZSTRIDE + zswz2(117 + (rr >> 4), rr & 15)] = 0.0f;
    }
    __syncthreads();

    chunk_gemm<true, false>(w0r, K0, 9 * CK, 15, false, hi, boff, zs, acc);
  }
  __syncthreads();   // all waves done reading zs before it is reused

  // bias + relu; rows 0..63 -> ys (feed layer 1), rows 64..127 -> zs (reduce)
#pragma unroll
  for (int nb = 0; nb < NB; ++nb) {
#pragma unroll
    for (int j = 0; j < 8; ++j) {
      const int row = (wave << 4) + j + (hi << 3);   // C/D layout: M = j + 8*hi
      float v = acc[nb][j] + b0[row];
      v = v > 0.0f ? v : 0.0f;
      if (wave < 4) ys[nb * (H1 * D) + row * D + lo16] = v;
      else          zs[nb * (H1 * D) + (row - H1) * D + lo16] = v;
      acc[nb][j] = 0.0f;                             // reset for layer 1
    }
  }
  __syncthreads();

  // direct0: out[b, r] = sum_d relu(y0[64+r, d])   (4 batches x 64 rows = 256)
  {
    const int nb = tid >> 6;
    const int r  = tid & 63;
    float sum = 0.0f;
#pragma unroll
    for (int d = 0; d < D; ++d) sum += zs[nb * (H1 * D) + r * D + d];
    out[(size_t)(bbase + nb) * 192 + r] = sum;
  }

  // ======================= layer 1 : K1 = 2496 ==========================
  for (int ch = 0; ch < 16; ++ch) {       // 16 chunks x 156 cols, 39 steps
    const int h0    = ch << 2;
    const int cbase = ch * CK;

    __syncthreads();
    // stage Z1 chunk: z[hh*39+mm, d] = y0relu[h0+hh][d] * x[mm][d]
    for (int nb = 0; nb < NB; ++nb) {
      const float* xb = xs + nb * (F * D);
      const float* yb = ys + nb * (H1 * D);
      float*       zb = zs + nb * ZSTRIDE;
      for (int hh = 0; hh < 4; ++hh) {
        for (int r = tid; r < F * D; r += TPB) {
          const int mm = r >> 4;
          const int d  = r & 15;
          zb[zswz2(hh * F + mm, d)] =
              yb[(h0 + hh) * D + d] * xb[mm * D + d];
        }
      }
    }
    __syncthreads();

    if (ch < 15)
      __builtin_prefetch(&w1r[cbase + CK + 2 * hi], 0, 0);

    chunk_gemm<false, true>(w1r, K1, cbase, 19, true, hi, boff, zs, acc);
  }
  __syncthreads();   // all waves done reading zs before reduce staging

  // bias + relu; stage full Y1 (128 rows) into zs for the d-reduction
#pragma unroll
  for (int nb = 0; nb < NB; ++nb) {
#pragma unroll
    for (int j = 0; j < 8; ++j) {
      const int row = (wave << 4) + j + (hi << 3);
      float v = acc[nb][j] + b1[row];
      v = v > 0.0f ? v : 0.0f;
      zs[nb * (L0N * D) + row * D + lo16] = v;
    }
  }
  __syncthreads();

  // direct1: out[b, 64+r] = sum_d relu(y1[r, d])  (4 x 128 = 512 sums)
#pragma unroll
  for (int rep = 0; rep < 2; ++rep) {
    const int idx = tid + rep * TPB;
    const int nb  = idx >> 7;
    const int r   = idx & 127;
    float sum = 0.0f;
#pragma unroll
    for (int d = 0; d < D; ++d) sum += zs[nb * (L0N * D) + r * D + d];
    out[(size_t)(bbase + nb) * 192 + 64 + r] = sum;
  }
}

extern "C" void kernel_launch(void* const* d_in, const int* in_sizes, int n_in,
                              void* d_out, int out_size, void* d_ws, size_t ws_size,
                              hipStream_t stream) {
  const float* x  = (const float*)d_in[0];
  const float* W0 = (const float*)d_in[1];
  const float* b0 = (const float*)d_in[2];
  const float* W1 = (const float*)d_in[3];
  const float* b1 = (const float*)d_in[4];
  float* out = (float*)d_out;

  dim3 grid(B_TOT / NB);   // 1024 blocks
  dim3 block(TPB);         // 8 wave32 waves
  hipLaunchKernelGGL(cin_wmma_kernel, grid, block, 0, stream,
                     x, W0, b0, W1, b1, out);
}